// phySenseCRF_55276229099888
// MI455X (gfx1250) — compile-verified
//
#include <hip/hip_runtime.h>

// Sizes fixed by the reference
#define B_  4
#define N_  48
#define S_  32
#define A_  16
#define AI_ 8
#define E_  512
#define K_  16

typedef __attribute__((ext_vector_type(2))) float v2f;
typedef __attribute__((ext_vector_type(4))) float v4f;
typedef __attribute__((ext_vector_type(8))) float v8f;
typedef __attribute__((ext_vector_type(2))) int   v2i;

#define LDS_AS __attribute__((address_space(3)))
#define GLB_AS __attribute__((address_space(1)))

#if __has_builtin(__builtin_amdgcn_global_load_async_to_lds_b64) && \
    __has_builtin(__builtin_amdgcn_s_wait_asynccnt)
#define USE_ASYNC 1
#else
#define USE_ASYNC 0
#endif

// ---------------------------------------------------------------------------
// Kernel 1: weighted unaries, beam selection (target forced to slot 0),
// unary gold log-softmax term.  One wave32 per (b,n); lane == label s.
// ---------------------------------------------------------------------------
__global__ __launch_bounds__(256) void unary_beam_kernel(
    const float* __restrict__ unaries,
    const float* __restrict__ behaviors,
    const float* __restrict__ wpu,
    const unsigned char* __restrict__ masks,
    const unsigned char* __restrict__ bmask,
    const int* __restrict__ targets,
    int* __restrict__ beam,
    float* __restrict__ goldu)
{
  const float INF = __builtin_inff();
  const int lane = threadIdx.x & 31;
  const int wid  = blockIdx.x * (blockDim.x >> 5) + (threadIdx.x >> 5);
  if (wid >= B_ * N_) return;                // uniform per-wave
  const int b = wid / N_;
  const int n = wid % N_;
  const int s = lane;

  const float mval = (float)masks[b * N_ + n];
  const float u    = unaries[(b * N_ + n) * S_ + s];

  // masked behavior mean over A; bool rows are 0/1 bytes -> sum == popcount
  const unsigned long long m0 = *(const unsigned long long*)(bmask + s * A_);
  const unsigned long long m1 = *(const unsigned long long*)(bmask + s * A_ + 8);
  const float* bh = behaviors + ((b * N_ + n) * S_ + s) * A_;
  float bs = 0.f;
#pragma unroll
  for (int a = 0; a < 8; ++a) {
    bs += ((m0 >> (8 * a)) & 1ull) ? bh[a]     : 0.f;
    bs += ((m1 >> (8 * a)) & 1ull) ? bh[a + 8] : 0.f;
  }
  float ac = (float)(__popcll(m0) + __popcll(m1));
  if (ac == 0.f) ac = 1.f;
  const float bu = (bs / ac) * mval;
  const float wu = u * mval + wpu[s] * bu;

  // beam: rank of each label with target forced to +inf (rank 0)
  const int tgt   = targets[b * N_ + n];
  const float sel = (s == tgt) ? INF : wu;
  int rank = 0;
#pragma unroll
  for (int j = 0; j < 32; ++j) {
    const float vj = __shfl(sel, j, 32);
    rank += (vj > sel) || (vj == sel && j < s);
  }
  const bool kept = (rank < K_);
  if (kept) beam[(b * N_ + n) * K_ + rank] = s;   // slot 0 == target

  // log-softmax over the kept set, gold at slot 0 (= target's value)
  float mk = kept ? wu : -INF;
  for (int off = 16; off; off >>= 1) mk = fmaxf(mk, __shfl_xor(mk, off, 32));
  float se = kept ? __expf(wu - mk) : 0.f;
  for (int off = 16; off; off >>= 1) se += __shfl_xor(se, off, 32);
  const float wut = __shfl(wu, tgt, 32);
  if (lane == 0)
    goldu[b * N_ + n] = (mval != 0.f) ? (wut - (mk + __logf(se))) : 0.f;
}

// ---------------------------------------------------------------------------
// Kernel 2: per-edge 16x16 binary potentials via WMMA, gold log-softmax term.
// One wave32 per edge.  The masked Ai-reduction for a 16-row tile is the
// diagonal of X (16x8) * M^T (8x16): two accumulated v_wmma_f32_16x16x4_f32.
// A layout (ISA 7.12.2): lanes 0-15 hold K=0,1; lanes 16-31 hold K=2,3.
// Diagonal of C/D: rows 0-7 -> lanes 0-7 (VGPR=lane), rows 8-15 -> lanes
// 24-31 (VGPR=lane-24).
// Gathers for tile t+1 are double-buffered into LDS with
// GLOBAL_LOAD_ASYNC_TO_LDS_B64 and retired via S_WAIT_ASYNCCNT.
// ---------------------------------------------------------------------------
__global__ __launch_bounds__(256) void edge_bin_kernel(
    const float* __restrict__ interactions,
    const float* __restrict__ wpb,
    const unsigned char* __restrict__ imask,
    const int* __restrict__ edges,
    const unsigned char* __restrict__ binm,
    const int* __restrict__ beam,
    float* __restrict__ goldb)
{
#if USE_ASYNC
  __shared__ float stage[2][8][32][4];     // [buf][wave][lane][4 floats]
#endif
  const float INF = __builtin_inff();
  const int lane = threadIdx.x & 31;
  const int wv   = threadIdx.x >> 5;
  const int wid  = blockIdx.x * (blockDim.x >> 5) + wv;
  const int b = wid / E_;
  const int e = wid % E_;

  const int e0 = edges[(b * E_ + e) * 2 + 0];
  const int e1 = edges[(b * E_ + e) * 2 + 1];
  const int* bt1 = beam + (b * N_ + e0) * K_;
  const int* bt2 = beam + (b * N_ + e1) * K_;

  const int rr   = lane & 15;               // tile row this lane feeds (A/B)
  const int s2   = bt2[rr];
  const int bt1l = bt1[rr];                 // lane-resident; broadcast by shfl
  const int cb   = (lane < 16) ? 0 : 2;     // K-pair base within a chunk
  const long ib  = ((long)((b * N_ + e0) * N_ + e1)) * (S_ * S_ * AI_);

  const bool own  = (lane < 8) || (lane >= 24);      // owns a diagonal element
  const int  vidx = (lane < 8) ? lane : (lane - 24); // VGPR index of diag
  const int  rd   = (lane < 8) ? lane : (lane - 16); // diag row 0..15
  const int  s2d  = bt2[rd];

  float lm = -INF, ls = 0.f, phi00 = 0.f;   // online logsumexp state

#if USE_ASYNC
  const GLB_AS float* gsrc = (const GLB_AS float*)interactions;
  {  // prologue: stage tile 0
    const int s1p = __shfl(bt1l, 0, 32);
    const long offp = ib + (long)((s1p * S_ + s2) * AI_ + cb);
    LDS_AS float* d = (LDS_AS float*)&stage[0][wv][lane][0];
    __builtin_amdgcn_global_load_async_to_lds_b64(
        (GLB_AS v2i*)(gsrc + offp),     (LDS_AS v2i*)(d),     0, 0);
    __builtin_amdgcn_global_load_async_to_lds_b64(
        (GLB_AS v2i*)(gsrc + offp + 4), (LDS_AS v2i*)(d + 2), 0, 0);
  }
#endif

  for (int t = 0; t < K_; ++t) {
    const int s1 = __shfl(bt1l, t, 32);
    const int ro = (s1 * S_ + s2) * AI_;
    v2f a0, a1, b0, b1;

#if USE_ASYNC
    if (t + 1 < K_) {                       // stage tile t+1, retire tile t
      const int s1n = __shfl(bt1l, t + 1, 32);
      const long offn = ib + (long)((s1n * S_ + s2) * AI_ + cb);
      LDS_AS float* d = (LDS_AS float*)&stage[(t + 1) & 1][wv][lane][0];
      __builtin_amdgcn_global_load_async_to_lds_b64(
          (GLB_AS v2i*)(gsrc + offn),     (LDS_AS v2i*)(d),     0, 0);
      __builtin_amdgcn_global_load_async_to_lds_b64(
          (GLB_AS v2i*)(gsrc + offn + 4), (LDS_AS v2i*)(d + 2), 0, 0);
      __builtin_amdgcn_s_wait_asynccnt(2);
    } else {
      __builtin_amdgcn_s_wait_asynccnt(0);
    }
    const v4f xv = *(const v4f*)&stage[t & 1][wv][lane][0];  // ds_load_b128
    a0.x = xv.x; a0.y = xv.y;               // Ai chunk 0..3 (this lane's pair)
    a1.x = xv.z; a1.y = xv.w;               // Ai chunk 4..7
#else
    const float* xp = interactions + ib + ro + cb;
    a0.x = xp[0]; a0.y = xp[1];
    a1.x = xp[4]; a1.y = xp[5];
    const int s1n = __shfl(bt1l, (t + 1) & (K_ - 1), 32);
    __builtin_prefetch(interactions + ib + (s1n * S_ + s2) * AI_ + cb, 0, 0);
#endif

    b0.x = (float)imask[ro + cb + 0]; b0.y = (float)imask[ro + cb + 1];
    b1.x = (float)imask[ro + cb + 4]; b1.y = (float)imask[ro + cb + 5];

    v8f acc = {};
    acc = __builtin_amdgcn_wmma_f32_16x16x4_f32(false, a0, false, b0,
                                                (short)0, acc, false, false);
    acc = __builtin_amdgcn_wmma_f32_16x16x4_f32(false, a1, false, b1,
                                                (short)0, acc, false, false);

    // extract this lane's diagonal element D[r,r] = sum_ai X[r,ai]*M[r,ai]
    float dv = acc[0];
#pragma unroll
    for (int i = 1; i < 8; ++i) dv = (vidx == i) ? acc[i] : dv;

    if (own) {
      // icnt = sum of 8 bool bytes == popcount of the 8-byte row
      const unsigned long long mrow =
          *(const unsigned long long*)(imask + (s1 * S_ + s2d) * AI_);
      float ic = (float)__popcll(mrow);
      if (ic == 0.f) ic = 1.f;
      const float phi = (dv / ic) * wpb[s1 * S_ + s2d];
      if (t == 0 && lane == 0) phi00 = phi;  // bin_phis[0,0]
      if (phi > lm) { ls = ls * __expf(lm - phi) + 1.f; lm = phi; }
      else          { ls += __expf(phi - lm); }
    }
  }

  // cross-lane logsumexp combine (-inf-safe)
  for (int off = 16; off; off >>= 1) {
    const float om = __shfl_xor(lm, off, 32);
    const float os = __shfl_xor(ls, off, 32);
    if (om != -INF) {
      if (lm == -INF)   { lm = om; ls = os; }
      else if (om > lm) { ls = ls * __expf(lm - om) + os; lm = om; }
      else              { ls += os * __expf(om - lm); }
    }
  }

  if (lane == 0)
    goldb[b * E_ + e] = binm[b * E_ + e] ? (phi00 - (lm + __logf(ls))) : 0.f;
}

// ---------------------------------------------------------------------------
// Kernel 3: deterministic fixed-order final reduction -> scalar nll.
// ---------------------------------------------------------------------------
__global__ void finalize_kernel(const float* __restrict__ goldu,
                                const float* __restrict__ goldb,
                                const unsigned char* __restrict__ masks,
                                float* __restrict__ out)
{
  const int lane = threadIdx.x;
  float r = 0.f;
  if (lane < B_) {
    float pll = 0.f, nb = 0.f;
    for (int n = 0; n < N_; ++n) pll += goldu[lane * N_ + n];
    for (int e = 0; e < E_; ++e) pll += goldb[lane * E_ + e];
    for (int n = 0; n < N_; ++n) nb += (float)masks[lane * N_ + n];
    r = pll / nb;
  }
  for (int off = 16; off; off >>= 1) r += __shfl_xor(r, off, 32);
  if (lane == 0) out[0] = -r / (float)B_;
}

// ---------------------------------------------------------------------------
extern "C" void kernel_launch(void* const* d_in, const int* in_sizes, int n_in,
                              void* d_out, int out_size, void* d_ws, size_t ws_size,
                              hipStream_t stream)
{
  const float* unaries      = (const float*)d_in[0];
  const float* behaviors    = (const float*)d_in[1];
  const float* interactions = (const float*)d_in[2];
  const float* wpu          = (const float*)d_in[3];
  const float* wpb          = (const float*)d_in[4];
  const unsigned char* masks = (const unsigned char*)d_in[5];   // bool
  const unsigned char* bmask = (const unsigned char*)d_in[6];   // bool
  const unsigned char* imask = (const unsigned char*)d_in[7];   // bool
  const int* edges           = (const int*)d_in[8];
  const unsigned char* binm  = (const unsigned char*)d_in[9];   // bool
  const int* targets         = (const int*)d_in[10];
  float* out = (float*)d_out;

  char* ws   = (char*)d_ws;
  int*   beam  = (int*)ws;                                        // B*N*K ints
  float* goldu = (float*)(ws + (size_t)B_ * N_ * K_ * sizeof(int)); // B*N
  float* goldb = goldu + B_ * N_;                                  // B*E

  unary_beam_kernel<<<(B_ * N_ + 7) / 8, 256, 0, stream>>>(
      unaries, behaviors, wpu, masks, bmask, targets, beam, goldu);
  edge_bin_kernel<<<(B_ * E_ + 7) / 8, 256, 0, stream>>>(
      interactions, wpb, imask, edges, binm, beam, goldb);
  finalize_kernel<<<1, 32, 0, stream>>>(goldu, goldb, masks, out);
}